// GCN_7103875907990
// MI455X (gfx1250) — compile-verified
//
#include <hip/hip_runtime.h>

typedef float v2f __attribute__((ext_vector_type(2)));
typedef float v8f __attribute__((ext_vector_type(8)));

constexpr int NN  = 10000;   // nodes per graph
constexpr int FIN = 8;       // in features
constexpr int CC  = 16;      // conv channels
constexpr int BB  = 16;      // batch
constexpr int EE  = 320000;  // edges
constexpr int KK  = 4000;    // top-k
constexpr int H1  = 256, H2 = 64;
constexpr int KC  = KK * CC; // 64000 = FC1 K dim
constexpr int SORT_M = 16384;

// ---------------- conv1 neighbor scatter-add (F_IN=8) ----------------
__global__ void scatter_add8(const int* __restrict__ ei, const float* __restrict__ x,
                             float* __restrict__ agg) {
  int t = blockIdx.x * 256 + threadIdx.x;
  if (t >= BB * EE) return;
  int b = t / EE, e = t - b * EE;
  int s = ei[e];        // src row of edge_index
  int d = ei[EE + e];   // dst row
  const float* xs = x + ((size_t)b * NN + s) * FIN;
  float* ad = agg + ((size_t)b * NN + d) * FIN;
#pragma unroll
  for (int f = 0; f < FIN; ++f) atomicAdd(ad + f, xs[f]);
}

// ---------------- conv1 node transform: h = relu(agg@W1n^T + x@W1r^T + b1) ----
__global__ void conv1_node(const float* __restrict__ x, const float* __restrict__ agg,
                           const float* __restrict__ W1n, const float* __restrict__ W1r,
                           const float* __restrict__ b1, float* __restrict__ h) {
  int t = blockIdx.x * 256 + threadIdx.x;
  if (t >= BB * NN) return;
  float xv[FIN], av[FIN];
  const float* xs = x + (size_t)t * FIN;
  const float* as = agg + (size_t)t * FIN;
#pragma unroll
  for (int f = 0; f < FIN; ++f) { xv[f] = xs[f]; av[f] = as[f]; }
  float* hd = h + (size_t)t * CC;
#pragma unroll
  for (int c = 0; c < CC; ++c) {
    float acc = b1[c];
#pragma unroll
    for (int f = 0; f < FIN; ++f)
      acc = fmaf(W1n[c * FIN + f], av[f], fmaf(W1r[c * FIN + f], xv[f], acc));
    hd[c] = fmaxf(acc, 0.0f);
  }
}

// ---------------- SAGPool neighbor scatter-add (C=16) ----------------
__global__ void scatter_add16(const int* __restrict__ ei, const float* __restrict__ h,
                              float* __restrict__ sagg) {
  int t = blockIdx.x * 256 + threadIdx.x;
  if (t >= BB * EE) return;
  int b = t / EE, e = t - b * EE;
  int s = ei[e];
  int d = ei[EE + e];
  const float* hs = h + ((size_t)b * NN + s) * CC;
  float* ad = sagg + ((size_t)b * NN + d) * CC;
#pragma unroll
  for (int c = 0; c < CC; ++c) atomicAdd(ad + c, hs[c]);
}

// ---------------- attention score = tanh(GraphConv(h) -> 1) -----------
__global__ void score_node(const float* __restrict__ h, const float* __restrict__ sagg,
                           const float* __restrict__ Wan, const float* __restrict__ War,
                           const float* __restrict__ ba, float* __restrict__ score) {
  int t = blockIdx.x * 256 + threadIdx.x;
  if (t >= BB * NN) return;
  const float* hs = h + (size_t)t * CC;
  const float* ss = sagg + (size_t)t * CC;
  float acc = ba[0];
#pragma unroll
  for (int c = 0; c < CC; ++c)
    acc = fmaf(Wan[c], ss[c], fmaf(War[c], hs[c], acc));
  score[t] = tanhf(acc);
}

// ---------------- per-graph full bitonic sort in 128KB LDS ------------
// key = (~orderedFloat(score) << 32) | index  -> ascending sort gives
// descending scores with lower-index tie break (matches jax.lax.top_k).
__global__ void __launch_bounds__(1024)
topk_kernel(const float* __restrict__ score, int* __restrict__ idx_out,
            float* __restrict__ vals_out) {
  extern __shared__ unsigned long long keys[];
  int b = blockIdx.x;
  for (int i = threadIdx.x; i < SORT_M; i += 1024) {
    unsigned long long key;
    if (i < NN) {
      unsigned u = __float_as_uint(score[b * NN + i]);
      unsigned m = (u & 0x80000000u) ? ~u : (u | 0x80000000u);  // asc-ordered
      key = ((unsigned long long)(~m) << 32) | (unsigned)i;      // desc scores
    } else {
      key = ~0ull;  // padding sinks to the end
    }
    keys[i] = key;
  }
  __syncthreads();
  for (int k = 2; k <= SORT_M; k <<= 1) {
    for (int j = k >> 1; j > 0; j >>= 1) {
      for (int i = threadIdx.x; i < SORT_M; i += 1024) {
        int ixj = i ^ j;
        if (ixj > i) {
          unsigned long long a = keys[i], c = keys[ixj];
          bool up = ((i & k) == 0);
          if ((a > c) == up) { keys[i] = c; keys[ixj] = a; }
        }
      }
      __syncthreads();
    }
  }
  for (int i = threadIdx.x; i < KK; i += 1024) {
    unsigned long long key = keys[i];
    unsigned d = (unsigned)(key >> 32);
    unsigned m = ~d;
    unsigned u = (m & 0x80000000u) ? (m ^ 0x80000000u) : ~m;   // invert mapping
    idx_out[b * KK + i]  = (int)(key & 0xFFFFFFFFu);
    vals_out[b * KK + i] = __uint_as_float(u);
  }
}

// ---------------- gather + scale: z[b,k,:] = h[b,idx,:] * val ---------
__global__ void gather_z(const float* __restrict__ h, const int* __restrict__ idxp,
                         const float* __restrict__ vals, float* __restrict__ z) {
  int t = blockIdx.x * 256 + threadIdx.x;
  if (t >= BB * KK) return;
  int b = t / KK;
  int n = idxp[t];
  float v = vals[t];
  const float4* hs = (const float4*)(h + ((size_t)b * NN + n) * CC);
  float4* zd = (float4*)(z + (size_t)t * CC);
#pragma unroll
  for (int j = 0; j < 4; ++j) {
    float4 q = hs[j];
    zd[j] = make_float4(q.x * v, q.y * v, q.z * v, q.w * v);
  }
}

// ---------------- FC1: (16 x 64000) @ (64000 x 256), split-K WMMA -----
// 512 waves total: ntile = wid&15 (N tile of 16), kchunk = wid>>4 (32 chunks
// of 500 WMMA K-steps). Partial 16x16 f32 tiles reduced with atomicAdd.
__global__ void __launch_bounds__(256)
fc1_wmma(const float* __restrict__ z, const float* __restrict__ W,
         float* __restrict__ acc) {
  int wid  = blockIdx.x * 8 + (threadIdx.x >> 5);
  int lane = threadIdx.x & 31;
  int ntile  = wid & 15;
  int kchunk = wid >> 4;                 // 0..31
  int kbase  = kchunk * (KC / 32);       // 2000 elements = 500 steps of 4
  int m   = lane & 15;
  int off = (lane >> 4) << 1;            // lanes 0-15: K{0,1}; 16-31: K{2,3}
  const float* ap = z + (size_t)m * KC + kbase + off;
  const float* bp = W + (size_t)(ntile * 16 + m) * KC + kbase + off;
  v8f c = {};
  for (int kk = 0; kk < KC / 32 / 4; ++kk) {   // 500 iterations
    v2f a  = *(const v2f*)ap;
    v2f bm = *(const v2f*)bp;
    c = __builtin_amdgcn_wmma_f32_16x16x4_f32(false, a, false, bm, (short)0, c,
                                              false, false);
    ap += 4;
    bp += 4;
  }
  int col   = ntile * 16 + (lane & 15);
  int rbase = (lane >> 4) << 3;
#pragma unroll
  for (int r = 0; r < 8; ++r)
    atomicAdd(&acc[(size_t)(rbase + r) * H1 + col], c[r]);
}

// ---------------- FC2: relu(fc1+b) @ Wfc2^T via WMMA, 4 waves ---------
__global__ void __launch_bounds__(128)
fc2_wmma(const float* __restrict__ fc1acc, const float* __restrict__ bfc1,
         const float* __restrict__ W2, float* __restrict__ fc2) {
  int wave = threadIdx.x >> 5;           // 0..3 -> N tile
  int lane = threadIdx.x & 31;
  int n0   = wave * 16;
  int m    = lane & 15;
  int off  = (lane >> 4) << 1;
  v8f c = {};
  for (int k = 0; k < H1; k += 4) {
    int ka = k + off;
    v2f a = { fmaxf(fc1acc[m * H1 + ka] + bfc1[ka], 0.0f),
              fmaxf(fc1acc[m * H1 + ka + 1] + bfc1[ka + 1], 0.0f) };
    v2f bm = *(const v2f*)(W2 + (size_t)(n0 + m) * H1 + ka);
    c = __builtin_amdgcn_wmma_f32_16x16x4_f32(false, a, false, bm, (short)0, c,
                                              false, false);
  }
  int col   = n0 + (lane & 15);
  int rbase = (lane >> 4) << 3;
#pragma unroll
  for (int r = 0; r < 8; ++r)
    fc2[(rbase + r) * H2 + col] = c[r];
}

// ---------------- FC3: relu(fc2+b) @ Wfc3^T + b3 ----------------------
__global__ void fc3_kernel(const float* __restrict__ fc2, const float* __restrict__ bfc2,
                           const float* __restrict__ W3, const float* __restrict__ bfc3,
                           float* __restrict__ out) {
  int b = threadIdx.x;
  if (b >= BB) return;
  float acc = bfc3[0];
#pragma unroll
  for (int o = 0; o < H2; ++o)
    acc = fmaf(fmaxf(fc2[b * H2 + o] + bfc2[o], 0.0f), W3[o], acc);
  out[b] = acc;
}

extern "C" void kernel_launch(void* const* d_in, const int* in_sizes, int n_in,
                              void* d_out, int out_size, void* d_ws, size_t ws_size,
                              hipStream_t stream) {
  const float* data = (const float*)d_in[0];   // (B, N*F_IN)
  const int*   ei   = (const int*)d_in[1];     // (2, E)
  const float* W1n  = (const float*)d_in[2];
  const float* W1r  = (const float*)d_in[3];
  const float* b1   = (const float*)d_in[4];
  const float* Wan  = (const float*)d_in[5];
  const float* War  = (const float*)d_in[6];
  const float* ba   = (const float*)d_in[7];
  const float* Wfc1 = (const float*)d_in[8];
  const float* bfc1 = (const float*)d_in[9];
  const float* Wfc2 = (const float*)d_in[10];
  const float* bfc2 = (const float*)d_in[11];
  const float* Wfc3 = (const float*)d_in[12];
  const float* bfc3 = (const float*)d_in[13];
  float* out = (float*)d_out;

  // workspace layout (floats)
  float* ws     = (float*)d_ws;
  float* agg    = ws;                                  // B*N*FIN   = 1,280,000
  float* h      = agg   + (size_t)BB * NN * FIN;       // B*N*C     = 2,560,000
  float* sagg   = h     + (size_t)BB * NN * CC;        // B*N*C     = 2,560,000
  float* score  = sagg  + (size_t)BB * NN * CC;        // B*N       =   160,000
  float* vals   = score + (size_t)BB * NN;             // B*K       =    64,000
  int*   idxp   = (int*)(vals + (size_t)BB * KK);      // B*K ints
  float* z      = (float*)(idxp + (size_t)BB * KK);    // B*K*C     = 1,024,000
  float* fc1    = z     + (size_t)BB * KC;             // B*H1      =     4,096
  float* fc2    = fc1   + (size_t)BB * H1;             // B*H2      =     1,024

  // zero the accumulation buffers
  hipMemsetAsync(agg,  0, (size_t)BB * NN * FIN * sizeof(float), stream);
  hipMemsetAsync(sagg, 0, (size_t)BB * NN * CC * sizeof(float), stream);
  hipMemsetAsync(fc1,  0, (size_t)BB * H1 * sizeof(float), stream);

  // conv1
  scatter_add8<<<(BB * EE + 255) / 256, 256, 0, stream>>>(ei, data, agg);
  conv1_node<<<(BB * NN + 255) / 256, 256, 0, stream>>>(data, agg, W1n, W1r, b1, h);

  // SAGPool scoring
  scatter_add16<<<(BB * EE + 255) / 256, 256, 0, stream>>>(ei, h, sagg);
  score_node<<<(BB * NN + 255) / 256, 256, 0, stream>>>(h, sagg, Wan, War, ba, score);

  // top-k sort per graph (128KB dynamic LDS; CDNA5 WGP has 320KB)
  hipFuncSetAttribute(reinterpret_cast<const void*>(topk_kernel),
                      hipFuncAttributeMaxDynamicSharedMemorySize,
                      SORT_M * (int)sizeof(unsigned long long));
  topk_kernel<<<BB, 1024, SORT_M * sizeof(unsigned long long), stream>>>(
      score, idxp, vals);

  // gather + scale
  gather_z<<<(BB * KK + 255) / 256, 256, 0, stream>>>(h, idxp, vals, z);

  // FC head: WMMA f32 16x16x4
  fc1_wmma<<<64, 256, 0, stream>>>(z, Wfc1, fc1);
  fc2_wmma<<<1, 128, 0, stream>>>(fc1, bfc1, Wfc2, fc2);
  fc3_kernel<<<1, 32, 0, stream>>>(fc2, bfc2, Wfc3, bfc3, out);
}